// GATLayer_6296422056075
// MI455X (gfx1250) — compile-verified
//
#include <hip/hip_runtime.h>
#include <hip/hip_bf16.h>

// ---------------------------------------------------------------------------
// GAT layer for MI455X (gfx1250, wave32).
//   K1: feat @ W_n via V_WMMA_F32_16X16X4_F32 (LDS-staged A tile, padded)
//   K2: el/er per-node einsum (attn_l/attn_r staged in LDS)
//   K3: per-edge logits + leaky relu + atomic segment-max (uint-encoded)
//   K4: exp(e - max[dst]) + atomic segment-sum
//   K5: wave-per-edge scatter: out[dst] += feat_src[src] * a   (coalesced atomics)
// ---------------------------------------------------------------------------

#define GAT_N 50000
#define GAT_E 800000
#define GAT_IN 256
#define GAT_H 8
#define GAT_D 32
#define GAT_HD 256   // H*D
#define GAT_T 4
#define GAT_F 16
#define NEG_SLOPE 0.2f

typedef float v2f __attribute__((ext_vector_type(2)));
typedef float v8f __attribute__((ext_vector_type(8)));

// order-preserving float<->uint encoding for atomicMax-based segment max
__device__ __forceinline__ unsigned int enc_f32(float f) {
    unsigned int u = __float_as_uint(f);
    return (u & 0x80000000u) ? ~u : (u | 0x80000000u);
}
__device__ __forceinline__ float dec_f32(unsigned int u) {
    return (u & 0x80000000u) ? __uint_as_float(u & 0x7fffffffu)
                             : __uint_as_float(~u);
}

// ---------------------------------------------------------------------------
// K1: feat_src[N,256] = feat[N,256] @ W_n[256,256]
// Block: 128 threads (4 waves). Each wave -> one 16x16 output tile.
// Block tile: rows [16), cols [64). A tile staged in LDS, stride 260 floats
// (260 % 64 == 4 -> conflict-free half-wave reads).
// ---------------------------------------------------------------------------
__global__ void __launch_bounds__(128)
gat_proj_wmma(const float* __restrict__ feat,
              const float* __restrict__ Wn,
              float* __restrict__ feat_src) {
    __shared__ float As[16 * 260];

    const int tid   = threadIdx.x;
    const int row0  = blockIdx.x * 16;
    const int wave  = tid >> 5;
    const int lane  = tid & 31;
    const int lhalf = lane & 15;
    const int hi    = (lane >> 4) & 1;          // 0: lanes 0-15, 1: lanes 16-31
    const int col0  = blockIdx.y * 64 + wave * 16;

    // cooperative, coalesced load of the 16x256 A tile
    for (int idx = tid; idx < 16 * 256; idx += 128) {
        int r = idx >> 8;
        int k = idx & 255;
        As[r * 260 + k] = feat[(row0 + r) * GAT_IN + k];
    }
    __syncthreads();

    v8f c = {0.f, 0.f, 0.f, 0.f, 0.f, 0.f, 0.f, 0.f};

    // A 16x4 fp32 layout: lanes 0-15 hold {K=k0,k0+1}, lanes 16-31 {k0+2,k0+3}
    // B  4x16 fp32 layout: symmetric (rows K striped across lanes / half-waves)
    #pragma unroll 4
    for (int k0 = 0; k0 < GAT_IN; k0 += 4) {
        const int kb = k0 + (hi ? 2 : 0);
        v2f av;
        av.x = As[lhalf * 260 + kb];
        av.y = As[lhalf * 260 + kb + 1];
        v2f bv;
        bv.x = Wn[kb * GAT_HD + col0 + lhalf];
        bv.y = Wn[(kb + 1) * GAT_HD + col0 + lhalf];
        c = __builtin_amdgcn_wmma_f32_16x16x4_f32(
                /*neg_a=*/false, av, /*neg_b=*/false, bv,
                /*c_mod=*/(short)0, c, /*reuse_a=*/false, /*reuse_b=*/false);
    }

    // C/D layout: VGPR r -> M = r + (hi?8:0), N = lhalf
    #pragma unroll
    for (int r = 0; r < 8; ++r) {
        int m = row0 + r + (hi ? 8 : 0);
        feat_src[m * GAT_HD + col0 + lhalf] = c[r];
    }
}

// ---------------------------------------------------------------------------
// K2: el[n,h,t] = sum_d feat_src[n,h*32+d] * attn_l[h,d,t]  (and er)
// one thread per (n,h); attn tables staged in LDS (2 x 1024 floats)
// ---------------------------------------------------------------------------
__global__ void __launch_bounds__(256)
gat_node_attn(const float* __restrict__ feat_src,
              const float* __restrict__ attn_l,
              const float* __restrict__ attn_r,
              float* __restrict__ el,
              float* __restrict__ er) {
    __shared__ float Al[GAT_H * GAT_D * GAT_T];   // 1024
    __shared__ float Ar[GAT_H * GAT_D * GAT_T];   // 1024
    for (int i = threadIdx.x; i < GAT_H * GAT_D * GAT_T; i += 256) {
        Al[i] = attn_l[i];
        Ar[i] = attn_r[i];
    }
    __syncthreads();

    int g = blockIdx.x * blockDim.x + threadIdx.x;
    if (g >= GAT_N * GAT_H) return;
    int n = g >> 3;
    int h = g & 7;

    float accl[GAT_T] = {0.f, 0.f, 0.f, 0.f};
    float accr[GAT_T] = {0.f, 0.f, 0.f, 0.f};
    const float* frow = feat_src + n * GAT_HD + h * GAT_D;
    #pragma unroll 4
    for (int d = 0; d < GAT_D; ++d) {
        float f = frow[d];
        #pragma unroll
        for (int t = 0; t < GAT_T; ++t) {
            accl[t] = fmaf(f, Al[(h * GAT_D + d) * GAT_T + t], accl[t]);
            accr[t] = fmaf(f, Ar[(h * GAT_D + d) * GAT_T + t], accr[t]);
        }
    }
    #pragma unroll
    for (int t = 0; t < GAT_T; ++t) {
        el[g * GAT_T + t] = accl[t];
        er[g * GAT_T + t] = accr[t];
    }
}

// ---------------------------------------------------------------------------
// K3: per (e,h): em = e_pro[e]·attn_m[h,:,t]; combine with mask; leaky relu;
// atomic segment-max into n_max[dst,h] (uint-encoded)
// ---------------------------------------------------------------------------
__global__ void __launch_bounds__(256)
gat_edge_logits(const float* __restrict__ e_pro,
                const float* __restrict__ mask,
                const float* __restrict__ attn_m,
                const float* __restrict__ el,
                const float* __restrict__ er,
                const int* __restrict__ src,
                const int* __restrict__ dst,
                float* __restrict__ e_arr,
                unsigned int* __restrict__ n_max) {
    __shared__ float Am[GAT_H * GAT_F * GAT_T];   // 512
    for (int i = threadIdx.x; i < GAT_H * GAT_F * GAT_T; i += 256)
        Am[i] = attn_m[i];
    __syncthreads();

    int g = blockIdx.x * blockDim.x + threadIdx.x;
    if (g >= GAT_E * GAT_H) return;
    int e = g >> 3;
    int h = g & 7;

    float m[GAT_T];
    #pragma unroll
    for (int t = 0; t < GAT_T; ++t) m[t] = mask[e * GAT_T + t];

    float em[GAT_T] = {0.f, 0.f, 0.f, 0.f};
    #pragma unroll
    for (int f = 0; f < GAT_F; ++f) {
        float p = e_pro[e * GAT_F + f];
        #pragma unroll
        for (int t = 0; t < GAT_T; ++t)
            em[t] = fmaf(p, Am[(h * GAT_F + f) * GAT_T + t], em[t]);
    }

    int s = src[e];
    int d = dst[e];
    float left = 0.f, right = 0.f, eme = 0.f;
    #pragma unroll
    for (int t = 0; t < GAT_T; ++t) {
        left  = fmaf(el[(s * GAT_H + h) * GAT_T + t], m[t], left);
        right = fmaf(er[(d * GAT_H + h) * GAT_T + t], m[t], right);
        eme   = fmaf(em[t], m[t], eme);
    }
    float v = (left + right + eme) * (1.0f / 3.0f);
    v = (v > 0.f) ? v : NEG_SLOPE * v;
    e_arr[g] = v;
    atomicMax(&n_max[d * GAT_H + h], enc_f32(v));
}

// ---------------------------------------------------------------------------
// K4: e_exp = exp(e - max[dst]); atomic segment-sum into n_sum[dst,h]
// ---------------------------------------------------------------------------
__global__ void __launch_bounds__(256)
gat_edge_exp(const int* __restrict__ dst,
             const unsigned int* __restrict__ n_max,
             float* __restrict__ e_arr,          // in: e, out: e_exp
             float* __restrict__ n_sum) {
    int g = blockIdx.x * blockDim.x + threadIdx.x;
    if (g >= GAT_E * GAT_H) return;
    int e = g >> 3;
    int h = g & 7;
    int d = dst[e];
    float mx = dec_f32(n_max[d * GAT_H + h]);
    float ex = __expf(e_arr[g] - mx);
    e_arr[g] = ex;
    atomicAdd(&n_sum[d * GAT_H + h], ex);
}

// ---------------------------------------------------------------------------
// K5: wave per edge; lane = d. a = e_exp/sum[dst]; out[dst] += feat_src[src]*a
// 32-lane contiguous atomics -> each atomic instruction covers 128B.
// ---------------------------------------------------------------------------
__global__ void __launch_bounds__(256)
gat_scatter(const float* __restrict__ feat_src,
            const float* __restrict__ e_exp,
            const float* __restrict__ n_sum,
            const int* __restrict__ src,
            const int* __restrict__ dst,
            float* __restrict__ out) {
    int wave = (blockIdx.x * blockDim.x + threadIdx.x) >> 5;
    if (wave >= GAT_E) return;
    int lane = threadIdx.x & 31;
    int s = src[wave];
    int d = dst[wave];
    const float* frow = feat_src + s * GAT_HD;
    float* orow = out + d * GAT_HD;
    #pragma unroll
    for (int h = 0; h < GAT_H; ++h) {
        float a = e_exp[wave * GAT_H + h] / n_sum[d * GAT_H + h];
        atomicAdd(&orow[h * GAT_D + lane], frow[h * GAT_D + lane] * a);
    }
}

// ---------------------------------------------------------------------------
extern "C" void kernel_launch(void* const* d_in, const int* in_sizes, int n_in,
                              void* d_out, int out_size, void* d_ws, size_t ws_size,
                              hipStream_t stream) {
    (void)in_sizes; (void)n_in; (void)ws_size;

    const float* feat   = (const float*)d_in[0];   // [N,256]
    const float* e_pro  = (const float*)d_in[1];   // [E,16]
    const float* mask   = (const float*)d_in[2];   // [E,4]
    const float* Wn     = (const float*)d_in[3];   // [256,256]
    const float* attn_l = (const float*)d_in[4];   // [8,32,4]
    const float* attn_r = (const float*)d_in[5];   // [8,32,4]
    const float* attn_m = (const float*)d_in[6];   // [8,16,4]
    const int*   src    = (const int*)d_in[7];     // [E]
    const int*   dst    = (const int*)d_in[8];     // [E]
    float* out = (float*)d_out;                    // [N,256]

    // workspace carve-up (bytes): ~92.8 MB total
    char* ws = (char*)d_ws;
    size_t off = 0;
    float* feat_src = (float*)(ws + off); off += (size_t)GAT_N * GAT_HD * 4;          // 51.2 MB
    float* el       = (float*)(ws + off); off += (size_t)GAT_N * GAT_H * GAT_T * 4;   //  6.4 MB
    float* er       = (float*)(ws + off); off += (size_t)GAT_N * GAT_H * GAT_T * 4;   //  6.4 MB
    float* e_arr    = (float*)(ws + off); off += (size_t)GAT_E * GAT_H * 4;           // 25.6 MB
    unsigned int* n_max = (unsigned int*)(ws + off); off += (size_t)GAT_N * GAT_H * 4;//  1.6 MB
    float* n_sum    = (float*)(ws + off); off += (size_t)GAT_N * GAT_H * 4;           //  1.6 MB

    // zero accumulators (graph-capture legal)
    hipMemsetAsync(out,   0, (size_t)out_size * sizeof(float), stream);
    hipMemsetAsync(n_max, 0, (size_t)GAT_N * GAT_H * 4, stream);  // enc(-inf) floor
    hipMemsetAsync(n_sum, 0, (size_t)GAT_N * GAT_H * 4, stream);

    // K1: projection GEMM (N=50000 = 3125*16 exactly; cols 256 = 4*64)
    dim3 g1(GAT_N / 16, GAT_HD / 64);
    gat_proj_wmma<<<g1, 128, 0, stream>>>(feat, Wn, feat_src);

    // K2: node attention logits
    int nh = GAT_N * GAT_H;
    gat_node_attn<<<(nh + 255) / 256, 256, 0, stream>>>(feat_src, attn_l, attn_r, el, er);

    // K3: edge logits + leaky relu + segment max
    int eh = GAT_E * GAT_H;
    gat_edge_logits<<<(eh + 255) / 256, 256, 0, stream>>>(
        e_pro, mask, attn_m, el, er, src, dst, e_arr, n_max);

    // K4: exp + segment sum
    gat_edge_exp<<<(eh + 255) / 256, 256, 0, stream>>>(dst, n_max, e_arr, n_sum);

    // K5: normalize + scatter-add aggregation (wave per edge)
    long long threads = (long long)GAT_E * 32;
    gat_scatter<<<(int)((threads + 255) / 256), 256, 0, stream>>>(
        feat_src, e_arr, n_sum, src, dst, out);
}